// BasicRGCN_25391846653982
// MI455X (gfx1250) — compile-verified
//
#include <hip/hip_runtime.h>
#include <hip/hip_bf16.h>

// ---------------------------------------------------------------------------
// Problem constants (match reference)
// ---------------------------------------------------------------------------
#define HIDDEN   128
#define IN_DIM   15
#define NUM_REL  4
#define N_NODES  100000
#define N_EDGES  2000000
#define N_GRAPHS 64
#define NSEG     (NUM_REL + 1)

// Workspace layout (in floats)
#define OFF_CNT   ((size_t)0)                              // [4*N] per (rel,node) edge count
#define OFF_AGG1  (OFF_CNT  + (size_t)NUM_REL*N_NODES)     // [4*N*16] layer-1 sums (K pad 15->16)
#define OFF_AGG2  (OFF_AGG1 + (size_t)NUM_REL*N_NODES*16)  // [4*N*128]
#define OFF_POOL  (OFF_AGG2 + (size_t)NUM_REL*N_NODES*128) // [64*128]
#define OFF_GCNT  (OFF_POOL + (size_t)N_GRAPHS*HIDDEN)     // [64]
#define ZERO_TOT  (OFF_GCNT + (size_t)N_GRAPHS)            // zero everything up to here
#define OFF_INV   (ZERO_TOT)                               // [4*N] 1/max(cnt,1)
#define OFF_H1    (OFF_INV  + (size_t)NUM_REL*N_NODES)     // [N*128] layer-1 output
#define OFF_XPAD  (OFF_H1   + (size_t)N_NODES*HIDDEN)      // [N*16] x padded K 15->16
#define OFF_BPAD1 (OFF_XPAD + (size_t)N_NODES*16)          // [5*16*128] packed layer-1 weights

typedef float v2f __attribute__((ext_vector_type(2)));
typedef float v8f __attribute__((ext_vector_type(8)));

__device__ __forceinline__ v8f wmma4(v2f a, v2f b, v8f c) {
    // D = A(16x4,f32) * B(4x16,f32) + C(16x16,f32)
    return __builtin_amdgcn_wmma_f32_16x16x4_f32(
        /*neg_a=*/false, a, /*neg_b=*/false, b,
        /*c_mod=*/(short)0, c, /*reuse_a=*/false, /*reuse_b=*/false);
}

// ---------------------------------------------------------------------------
// Zero the atomic-accumulator region of the workspace (float4 grid stride)
// ---------------------------------------------------------------------------
__global__ void k_zero(float4* __restrict__ p, long long n4) {
    long long i = (long long)blockIdx.x * blockDim.x + threadIdx.x;
    long long stride = (long long)gridDim.x * blockDim.x;
    float4 z = make_float4(0.f, 0.f, 0.f, 0.f);
    for (; i < n4; i += stride) p[i] = z;
}

// ---------------------------------------------------------------------------
// Stage x into K-padded layout [N][16] (slot 15 = 0)
// ---------------------------------------------------------------------------
__global__ void k_pad_x(const float* __restrict__ x, float* __restrict__ xpad) {
    long long t = (long long)blockIdx.x * blockDim.x + threadIdx.x;
    if (t >= (long long)N_NODES * 16) return;
    int n = (int)(t >> 4);
    int k = (int)(t & 15);
    xpad[t] = (k < IN_DIM) ? x[(size_t)n * IN_DIM + k] : 0.f;
}

// ---------------------------------------------------------------------------
// Pack [root1 ; W1_0..W1_3] into Bpad1 [5][16][128] with zero-padded K rows
// ---------------------------------------------------------------------------
__global__ void k_pad_w1(const float* __restrict__ root1,
                         const float* __restrict__ W1,
                         float* __restrict__ Bpad1) {
    int t = blockIdx.x * blockDim.x + threadIdx.x;
    if (t >= NSEG * 16 * HIDDEN) return;
    int seg = t / (16 * HIDDEN);
    int k   = (t / HIDDEN) & 15;
    int c   = t & (HIDDEN - 1);
    float v = 0.f;
    if (k < IN_DIM) {
        v = (seg == 0) ? root1[(size_t)k * HIDDEN + c]
                       : W1[((size_t)(seg - 1) * IN_DIM + k) * HIDDEN + c];
    }
    Bpad1[t] = v;
}

// ---------------------------------------------------------------------------
// Edge pass, layer 1: 16 threads/edge. f<15 scatter x[src][f] into agg1,
// the padding lane (f==15) accumulates the per-(rel,dst) edge count.
// ---------------------------------------------------------------------------
__global__ void k_edge_agg1(const float* __restrict__ x,
                            const int*   __restrict__ ei,
                            const int*   __restrict__ et,
                            float* __restrict__ agg1,
                            float* __restrict__ cnt) {
    long long t = (long long)blockIdx.x * blockDim.x + threadIdx.x;
    long long e = t >> 4;
    int f = (int)(t & 15);
    if (e >= N_EDGES) return;
    int src = ei[e];
    int dst = ei[(long long)N_EDGES + e];
    int r   = et[e];
    if (f < IN_DIM) {
        atomicAdd(&agg1[(((size_t)r * N_NODES) + dst) * 16 + f],
                  x[(size_t)src * IN_DIM + f]);
    } else {
        atomicAdd(&cnt[(size_t)r * N_NODES + dst], 1.0f);
    }
}

// ---------------------------------------------------------------------------
// inv = 1 / max(cnt, 1)
// ---------------------------------------------------------------------------
__global__ void k_inv(const float* __restrict__ cnt, float* __restrict__ inv, int n) {
    int i = blockIdx.x * blockDim.x + threadIdx.x;
    if (i < n) inv[i] = 1.0f / fmaxf(cnt[i], 1.0f);
}

// ---------------------------------------------------------------------------
// Layer 1 GEMM: h1 = relu( [x | mean_0..3] @ Bpad1 + b1 )
// One block = 16 nodes; wave w (threadIdx.y) owns columns [16w, 16w+16).
// All loads unconditional (padded operands) -> no EXEC juggling.
// ---------------------------------------------------------------------------
__global__ void __launch_bounds__(256)
k_layer1(const float* __restrict__ xpad,
         const float* __restrict__ Bpad1,
         const float* __restrict__ b1,
         const float* __restrict__ agg1,
         const float* __restrict__ inv,
         float* __restrict__ h1) {
    const int tileM = blockIdx.x;          // 0..6249
    const int wave  = threadIdx.y;         // 0..7
    const int lane  = threadIdx.x;         // 0..31
    const int half  = lane >> 4;           // K sub-select
    const int m     = lane & 15;
    const int row   = tileM * 16 + m;      // A-row (node)
    const int col   = wave * 16 + m;       // B/C column

    v8f acc = {0.f,0.f,0.f,0.f,0.f,0.f,0.f,0.f};

    #pragma unroll
    for (int seg = 0; seg < NSEG; ++seg) {
        const float* Ar;
        float s;
        if (seg == 0) {
            Ar = xpad + (size_t)row * 16;
            s  = 1.0f;
        } else {
            int r = seg - 1;
            Ar = agg1 + ((size_t)r * N_NODES + row) * 16;
            s  = inv[(size_t)r * N_NODES + row];
        }
        const float* Br = Bpad1 + (size_t)seg * 16 * HIDDEN;
        #pragma unroll
        for (int k0 = 0; k0 < 16; k0 += 4) {
            int ka = k0 + 2 * half;
            v2f a = *(const v2f*)(Ar + ka);     // global_load_b64
            a.x *= s; a.y *= s;
            v2f b;
            b.x = Br[(size_t)(ka    ) * HIDDEN + col];
            b.y = Br[(size_t)(ka + 1) * HIDDEN + col];
            acc = wmma4(a, b, acc);
        }
    }

    const float bias = b1[col];
    #pragma unroll
    for (int i = 0; i < 8; ++i) {
        int orow = tileM * 16 + half * 8 + i;   // C/D layout: vgpr i -> rows i / i+8
        float v = acc[i] + bias;
        h1[(size_t)orow * HIDDEN + col] = v > 0.f ? v : 0.f;
    }
}

// ---------------------------------------------------------------------------
// Edge pass, layer 2: 32 threads/edge, each moves 4 consecutive floats.
// Gather of h1[src] is one contiguous 512B line per edge (coalesced).
// ---------------------------------------------------------------------------
__global__ void k_edge_agg2(const float* __restrict__ h1,
                            const int*   __restrict__ ei,
                            const int*   __restrict__ et,
                            float* __restrict__ agg2) {
    long long t = (long long)blockIdx.x * blockDim.x + threadIdx.x;
    long long e = t >> 5;
    int q = (int)(t & 31);
    if (e >= N_EDGES) return;
    int src = ei[e];
    int dst = ei[(long long)N_EDGES + e];
    int r   = et[e];
    const float4 v = *(const float4*)(h1 + (size_t)src * HIDDEN + q * 4);
    float* d = agg2 + ((size_t)r * N_NODES + dst) * HIDDEN + q * 4;
    atomicAdd(d + 0, v.x);
    atomicAdd(d + 1, v.y);
    atomicAdd(d + 2, v.z);
    atomicAdd(d + 3, v.w);
}

// ---------------------------------------------------------------------------
// Layer 2 GEMM + ReLU, epilogue accumulates directly into per-graph pool.
// ---------------------------------------------------------------------------
__global__ void __launch_bounds__(256)
k_layer2(const float* __restrict__ h1,
         const float* __restrict__ root2,
         const float* __restrict__ W2,
         const float* __restrict__ b2,
         const float* __restrict__ agg2,
         const float* __restrict__ inv,
         const int*   __restrict__ batch,
         float* __restrict__ pool) {
    const int tileM = blockIdx.x;
    const int wave  = threadIdx.y;
    const int lane  = threadIdx.x;
    const int half  = lane >> 4;
    const int m     = lane & 15;
    const int row   = tileM * 16 + m;
    const int col   = wave * 16 + m;

    v8f acc = {0.f,0.f,0.f,0.f,0.f,0.f,0.f,0.f};

    for (int seg = 0; seg < NSEG; ++seg) {
        const float* Ar;
        const float* Br;
        float s;
        if (seg == 0) {
            Ar = h1 + (size_t)row * HIDDEN;
            Br = root2;
            s  = 1.0f;
        } else {
            int r = seg - 1;
            Ar = agg2 + ((size_t)r * N_NODES + row) * HIDDEN;
            Br = W2 + (size_t)r * HIDDEN * HIDDEN;
            s  = inv[(size_t)r * N_NODES + row];
        }
        #pragma unroll 4
        for (int k0 = 0; k0 < HIDDEN; k0 += 4) {
            int ka = k0 + 2 * half;
            v2f a = *(const v2f*)(Ar + ka);     // global_load_b64
            a.x *= s; a.y *= s;
            v2f b;
            b.x = Br[(size_t)(ka    ) * HIDDEN + col];
            b.y = Br[(size_t)(ka + 1) * HIDDEN + col];
            acc = wmma4(a, b, acc);
        }
    }

    const float bias = b2[col];
    #pragma unroll
    for (int i = 0; i < 8; ++i) {
        int orow = tileM * 16 + half * 8 + i;
        int g = batch[orow];
        float v = acc[i] + bias;
        v = v > 0.f ? v : 0.f;
        atomicAdd(&pool[(size_t)g * HIDDEN + col], v);
    }
}

// ---------------------------------------------------------------------------
// Per-graph node counts
// ---------------------------------------------------------------------------
__global__ void k_gcnt(const int* __restrict__ batch, float* __restrict__ gcnt) {
    int n = blockIdx.x * blockDim.x + threadIdx.x;
    if (n < N_NODES) atomicAdd(&gcnt[batch[n]], 1.0f);
}

// ---------------------------------------------------------------------------
// out = pool / max(gcnt, 1)
// ---------------------------------------------------------------------------
__global__ void k_final(const float* __restrict__ pool,
                        const float* __restrict__ gcnt,
                        float* __restrict__ out) {
    int i = blockIdx.x * blockDim.x + threadIdx.x;
    if (i < N_GRAPHS * HIDDEN) {
        int g = i >> 7;
        out[i] = pool[i] / fmaxf(gcnt[g], 1.0f);
    }
}

// ---------------------------------------------------------------------------
extern "C" void kernel_launch(void* const* d_in, const int* in_sizes, int n_in,
                              void* d_out, int out_size, void* d_ws, size_t ws_size,
                              hipStream_t stream) {
    const float* x     = (const float*)d_in[0];
    const float* W1    = (const float*)d_in[1];
    const float* root1 = (const float*)d_in[2];
    const float* b1    = (const float*)d_in[3];
    const float* W2    = (const float*)d_in[4];
    const float* root2 = (const float*)d_in[5];
    const float* b2    = (const float*)d_in[6];
    const int*   ei    = (const int*)d_in[7];
    const int*   et    = (const int*)d_in[8];
    const int*   batch = (const int*)d_in[9];
    float* out = (float*)d_out;

    float* ws    = (float*)d_ws;
    float* cnt   = ws + OFF_CNT;
    float* agg1  = ws + OFF_AGG1;
    float* agg2  = ws + OFF_AGG2;
    float* pool  = ws + OFF_POOL;
    float* gcnt  = ws + OFF_GCNT;
    float* inv   = ws + OFF_INV;
    float* h1    = ws + OFF_H1;
    float* xpad  = ws + OFF_XPAD;
    float* Bpad1 = ws + OFF_BPAD1;

    // 1. zero all atomic accumulators (cnt|agg1|agg2|pool|gcnt are contiguous)
    long long n4 = (long long)(ZERO_TOT / 4);
    k_zero<<<4096, 256, 0, stream>>>((float4*)ws, n4);

    // 2. stage padded operands
    {
        long long threads = (long long)N_NODES * 16;
        k_pad_x<<<(int)((threads + 255) / 256), 256, 0, stream>>>(x, xpad);
        k_pad_w1<<<(NSEG * 16 * HIDDEN + 255) / 256, 256, 0, stream>>>(root1, W1, Bpad1);
    }

    // 3. layer-1 edge aggregation (+ relation counts)
    {
        long long threads = (long long)N_EDGES * 16;
        k_edge_agg1<<<(int)((threads + 255) / 256), 256, 0, stream>>>(x, ei, et, agg1, cnt);
    }

    // 4. reciprocal counts
    {
        int n = NUM_REL * N_NODES;
        k_inv<<<(n + 255) / 256, 256, 0, stream>>>(cnt, inv, n);
    }

    // 5. layer-1 WMMA GEMM -> h1
    k_layer1<<<N_NODES / 16, dim3(32, 8), 0, stream>>>(xpad, Bpad1, b1, agg1, inv, h1);

    // 6. layer-2 edge aggregation
    {
        long long threads = (long long)N_EDGES * 32;
        k_edge_agg2<<<(int)((threads + 255) / 256), 256, 0, stream>>>(h1, ei, et, agg2);
    }

    // 7. per-graph counts (independent of layer-2 GEMM)
    k_gcnt<<<(N_NODES + 255) / 256, 256, 0, stream>>>(batch, gcnt);

    // 8. layer-2 WMMA GEMM + ReLU -> pooled sums
    k_layer2<<<N_NODES / 16, dim3(32, 8), 0, stream>>>(h1, root2, W2, b2, agg2, inv,
                                                       batch, pool);

    // 9. mean pool -> output
    k_final<<<(N_GRAPHS * HIDDEN + 255) / 256, 256, 0, stream>>>(pool, gcnt, out);
}